// LogProb_13554916786231
// MI455X (gfx1250) — compile-verified
//
#include <hip/hip_runtime.h>
#include <math.h>

typedef float v2f __attribute__((ext_vector_type(2)));
typedef float v8f __attribute__((ext_vector_type(8)));

#define APAD 20    // stride*4 mod 64 hits all banks; rows 16B-aligned
#define BPAD 36
#define HPAD 260

#if defined(__has_builtin)
#if __has_builtin(__builtin_amdgcn_global_load_async_to_lds_b128) && \
    __has_builtin(__builtin_amdgcn_s_wait_asynccnt)
#define HAVE_ASYNC_LDS 1
#endif
#endif

#ifdef HAVE_ASYNC_LDS
typedef int vsi4 __attribute__((__vector_size__(4 * sizeof(int))));
typedef __attribute__((address_space(1))) vsi4* g4ptr_t;
typedef __attribute__((address_space(3))) vsi4* l4ptr_t;
__device__ __forceinline__ void async_b128(const float* g, float* l) {
  __builtin_amdgcn_global_load_async_to_lds_b128(
      (g4ptr_t)(void*)g, (l4ptr_t)(void*)l, 0, 0);
}
#endif

__device__ __forceinline__ v8f wmma4(v2f a, v2f b, v8f c) {
  // V_WMMA_F32_16X16X4_F32 : D = A(16x4,f32) * B(4x16,f32) + C(16x16,f32)
  return __builtin_amdgcn_wmma_f32_16x16x4_f32(false, a, false, b, (short)0, c, false, false);
}

__device__ __forceinline__ float ldg_guard(const float* __restrict__ p, long long idx, bool ok) {
  float v = p[ok ? idx : 0];     // clamped address: branchless load + select
  return ok ? v : 0.0f;
}

__device__ __forceinline__ float4 ldg4_guard(const float* __restrict__ p, long long idx, bool ok) {
  float4 v = *(const float4*)&p[ok ? idx : 0];
  float4 z = {0.f, 0.f, 0.f, 0.f};
  return ok ? v : z;
}

// ---------------- small utility kernels ----------------
__global__ void k_zero(float* p, int n) {
  int i = blockIdx.x * blockDim.x + threadIdx.x;
  if (i < n) p[i] = 0.0f;
}

__global__ void k_ildj(const float* __restrict__ M, int s, float hsq, float* ildj) {
  if (threadIdx.x == 0 && blockIdx.x == 0) {
    float tr = 0.0f;
    for (int i = 0; i < s; ++i) tr += M[(size_t)i * s + i];
    atomicAdd(ildj, expf(tr) * hsq);
  }
}

__global__ void k_expm_prep(const float* __restrict__ M, float* Asc, float* term,
                            float* outm, int s) {
  int i = blockIdx.x * blockDim.x + threadIdx.x;
  int n2 = s * s;
  if (i < n2) {
    Asc[i] = -M[i] * (1.0f / 64.0f);            // (-M) / 2^6
    float id = ((i / s) == (i % s)) ? 1.0f : 0.0f;
    term[i] = id;
    outm[i] = id;
  }
}

__global__ void k_accum4(float4* term, float4* outm, const float4* __restrict__ tmp, int n4) {
  int i = blockIdx.x * blockDim.x + threadIdx.x;
  if (i < n4) {
    float4 t = tmp[i];
    term[i] = t;
    float4 o = outm[i];
    o.x += t.x; o.y += t.y; o.z += t.z; o.w += t.w;
    outm[i] = o;
  }
}

__global__ void k_copy4(float4* dst, const float4* __restrict__ src, int n4) {
  int i = blockIdx.x * blockDim.x + threadIdx.x;
  if (i < n4) dst[i] = src[i];
}

// ---------------- LDS-staged WMMA-f32 GEMM: C = alpha * A(n,n)@B(n,n) --------
// 4 waves / block, 32x32 macro-tile, K-chunk 16. Full tiles use async
// global->LDS b128 copies (ASYNCcnt path) when available.
__global__ void __launch_bounds__(128) k_gemm(const float* __restrict__ A,
                                              const float* __restrict__ B,
                                              float* __restrict__ C, int n, float alpha) {
  __shared__ float Ash[32 * APAD];
  __shared__ float Bsh[16 * BPAD];
  int tid = threadIdx.x;
  int lane = tid & 31, wave = tid >> 5;
  int l15 = lane & 15, lhi = lane >> 4;
  int wm = wave >> 1, wn = wave & 1;
  int mBase = blockIdx.y * 32, nBase = blockIdx.x * 32;
  v8f acc = {0.f, 0.f, 0.f, 0.f, 0.f, 0.f, 0.f, 0.f};

  // per-thread staging coordinates (n is always a multiple of 4)
  int am = tid >> 2, aq = (tid & 3) * 4;         // A: 32 rows x 4 float4
  int bk = tid >> 3, bq = (tid & 7) * 4;         // B: 16 rows x 8 float4

  for (int kb = 0; kb < n; kb += 16) {
    bool full = (mBase + 32 <= n) && (nBase + 32 <= n) && (kb + 16 <= n);
#ifdef HAVE_ASYNC_LDS
    if (full) {
      async_b128(&A[(size_t)(mBase + am) * n + kb + aq], &Ash[am * APAD + aq]);
      async_b128(&B[(size_t)(kb + bk) * n + nBase + bq], &Bsh[bk * BPAD + bq]);
      __builtin_amdgcn_s_wait_asynccnt(0);
    } else
#endif
    {
      bool okA = (mBase + am < n) && (kb + aq < n);
      *(float4*)&Ash[am * APAD + aq] =
          ldg4_guard(A, (long long)(mBase + am) * n + kb + aq, okA);
      bool okB = (kb + bk < n) && (nBase + bq < n);
      *(float4*)&Bsh[bk * BPAD + bq] =
          ldg4_guard(B, (long long)(kb + bk) * n + nBase + bq, okB);
      (void)full;
    }
    __syncthreads();
    for (int ks = 0; ks < 4; ++ks) {
      int kk = ks * 4 + 2 * lhi;
      v2f a, b;
      a.x = Ash[(wm * 16 + l15) * APAD + kk];
      a.y = Ash[(wm * 16 + l15) * APAD + kk + 1];
      b.x = Bsh[kk * BPAD + wn * 16 + l15];
      b.y = Bsh[(kk + 1) * BPAD + wn * 16 + l15];
      acc = wmma4(a, b, acc);
    }
    __syncthreads();
  }
  int col = nBase + wn * 16 + l15;
  for (int r = 0; r < 8; ++r) {
    int m = mBase + wm * 16 + r + 8 * lhi;
    if (m < n && col < n) C[(size_t)m * n + col] = alpha * acc[r];
  }
}

// ---------------- level kernel A: fused s2d + E@(y-eb), LDS-staged -----------
__global__ void __launch_bounds__(128) k_mix(const float* __restrict__ Xin,
                                             float* __restrict__ Xout,
                                             const float* __restrict__ E,
                                             const float* __restrict__ eb,
                                             int B_, int H2, int W2, int Cin, int s) {
  __shared__ float Ash[32 * APAD];
  __shared__ float Bsh[16 * BPAD];
  int tid = threadIdx.x;
  int lane = tid & 31, wave = tid >> 5;
  int l15 = lane & 15, lhi = lane >> 4;
  int wm = wave >> 1, wn = wave & 1;
  int P = B_ * H2 * W2;
  int C4 = 4 * Cin;
  int pBase = blockIdx.x * 32, nBase = blockIdx.y * 32;
  v8f acc = {0.f, 0.f, 0.f, 0.f, 0.f, 0.f, 0.f, 0.f};

  for (int kb = 0; kb < s; kb += 16) {
    // stage A: s2d gather minus eb (branchless, clamped)
    for (int idx = tid; idx < 512; idx += 128) {
      int m = idx >> 4, k = idx & 15;
      int pp = pBase + m, j = kb + k;
      bool ok = (pp < P) && (j < s);
      int ppc = ok ? pp : 0, jc = ok ? j : 0;
      int w2 = ppc % W2; int t = ppc / W2; int h2 = t % H2; int b = t / H2;
      int a2 = jc / (2 * Cin); int rem = jc - a2 * 2 * Cin;
      int b2 = rem / Cin;      int c  = rem - b2 * Cin;
      float v = Xin[(((size_t)(b * 2 * H2 + 2 * h2 + a2)) * (2 * W2) + (2 * w2 + b2)) * Cin + c]
                - eb[jc];
      Ash[m * APAD + k] = ok ? v : 0.0f;
    }
    // stage B: B[j][i] = E[i][j] — vector load along E row, scatter to LDS col
    {
      int nn = tid >> 2, q = (tid & 3) * 4;      // 32 cols x 4 float4 rows
      int i2 = nBase + nn, j4 = kb + q;
      bool ok = (i2 < s) && (j4 < s);            // s % 4 == 0 -> exact check
      float4 v = ldg4_guard(E, (long long)(ok ? i2 : 0) * s + (ok ? j4 : 0), ok);
      Bsh[(q + 0) * BPAD + nn] = v.x;
      Bsh[(q + 1) * BPAD + nn] = v.y;
      Bsh[(q + 2) * BPAD + nn] = v.z;
      Bsh[(q + 3) * BPAD + nn] = v.w;
    }
    __syncthreads();
    for (int ks = 0; ks < 4; ++ks) {
      int kk = ks * 4 + 2 * lhi;
      v2f a, bv;
      a.x = Ash[(wm * 16 + l15) * APAD + kk];
      a.y = Ash[(wm * 16 + l15) * APAD + kk + 1];
      bv.x = Bsh[kk * BPAD + wn * 16 + l15];
      bv.y = Bsh[(kk + 1) * BPAD + wn * 16 + l15];
      acc = wmma4(a, bv, acc);
    }
    __syncthreads();
  }
  int ch = nBase + wn * 16 + l15;
  for (int r = 0; r < 8; ++r) {
    int pm = pBase + wm * 16 + r + 8 * lhi;
    if (pm < P && ch < s) Xout[(size_t)pm * C4 + ch] = acc[r];
  }
}

// ---------------- level kernel A2: passthrough channels [s, 4C) --------------
__global__ void k_passthru(const float* __restrict__ Xin, float* __restrict__ Xout,
                           int B_, int H2, int W2, int Cin, int s) {
  int C4 = 4 * Cin;
  int npass = C4 - s;
  long long total = (long long)B_ * H2 * W2 * npass;
  long long i = (long long)blockIdx.x * blockDim.x + threadIdx.x;
  if (i >= total) return;
  int ch = s + (int)(i % npass);
  long long p = i / npass;
  int w2 = (int)(p % W2); long long t = p / W2;
  int h2 = (int)(t % H2); int b = (int)(t / H2);
  int a2 = ch / (2 * Cin); int rem = ch - a2 * 2 * Cin;
  int b2 = rem / Cin;      int c  = rem - b2 * Cin;
  Xout[p * C4 + ch] =
      Xin[(((size_t)(b * 2 * H2 + 2 * h2 + a2)) * (2 * W2) + (2 * w2 + b2)) * Cin + c];
}

// ---------------- level kernel B: 3x3 conv (implicit GEMM) + projection ------
__global__ void __launch_bounds__(128) k_convproj(float* __restrict__ X,
                                                  const float* __restrict__ cw,
                                                  const float* __restrict__ cb,
                                                  const float* __restrict__ dw,
                                                  const float* __restrict__ db,
                                                  int B_, int H2, int W2, int C4, int d) {
  __shared__ float hbuf[32 * HPAD];   // 32 px x 256 h-channels (padded)
  __shared__ float Ash[32 * APAD];    // im2col chunk: 32 px x 16 k
  __shared__ float Bw[16 * HPAD];     // weight chunk: 16 k x 256 co (padded)
  int tid = threadIdx.x;
  int lane = tid & 31, wave = tid >> 5;
  int l15 = lane & 15, lhi = lane >> 4;
  int wm = wave >> 1, wn = wave & 1;
  int pBase = blockIdx.x * 32;
  int P = B_ * H2 * W2;
  int K = 9 * d;

  // ---- Phase 1: h = relu(im2col(x0) @ cw + cb) ----
  v8f acc1[8];
  for (int nc = 0; nc < 8; ++nc) acc1[nc] = (v8f){0.f, 0.f, 0.f, 0.f, 0.f, 0.f, 0.f, 0.f};

  for (int kb = 0; kb < K; kb += 16) {
    // stage A: im2col gather (branchless, clamped)
    for (int idx = tid; idx < 512; idx += 128) {
      int m = idx >> 4, k = idx & 15;
      int pp = pBase + m, kk = kb + k;
      bool ok = (pp < P) && (kk < K);
      int ppc = ok ? pp : 0, kkc = ok ? kk : 0;
      int w2 = ppc % W2; int t = ppc / W2; int h2 = t % H2; int b = t / H2;
      int tap = kkc / d, cc = kkc - tap * d;
      int hh = h2 + tap / 3 - 1, ww = w2 + tap % 3 - 1;
      bool inb = ok && (hh >= 0) && (hh < H2) && (ww >= 0) && (ww < W2);
      int hhc = inb ? hh : 0, wwc = inb ? ww : 0;
      float v = X[(((size_t)(b * H2 + hhc)) * W2 + wwc) * C4 + cc];
      Ash[m * APAD + k] = inb ? v : 0.0f;
    }
    // stage B: cw chunk 16 x 256, vectorized (co % 4 == 0)
    for (int idx = tid; idx < 1024; idx += 128) {
      int k = idx >> 6, co4 = (idx & 63) * 4;
      int kk = kb + k;
      bool ok = kk < K;
      float4 v = ldg4_guard(cw, (long long)(ok ? kk : 0) * 256 + co4, ok);
      *(float4*)&Bw[k * HPAD + co4] = v;
    }
    __syncthreads();
    for (int ks = 0; ks < 4; ++ks) {
      int kk = ks * 4 + 2 * lhi;
      v2f a;
      a.x = Ash[(wm * 16 + l15) * APAD + kk];
      a.y = Ash[(wm * 16 + l15) * APAD + kk + 1];
      for (int nc = 0; nc < 8; ++nc) {
        v2f bv;
        bv.x = Bw[kk * HPAD + nc * 32 + wn * 16 + l15];
        bv.y = Bw[(kk + 1) * HPAD + nc * 32 + wn * 16 + l15];
        acc1[nc] = wmma4(a, bv, acc1[nc]);
      }
    }
    __syncthreads();
  }
  // write h to LDS with bias + relu
  for (int nc = 0; nc < 8; ++nc) {
    int co = nc * 32 + wn * 16 + l15;
    float bias = cb[co];
    for (int r = 0; r < 8; ++r) {
      int m = wm * 16 + r + 8 * lhi;
      float h = acc1[nc][r] + bias;
      hbuf[m * HPAD + co] = h > 0.f ? h : 0.f;
    }
  }
  __syncthreads();

  // ---- Phase 2: shift = h @ dw; x1 -= shift + db (in place) ----
  int Jt = (d + 31) / 32;
  for (int jb = 0; jb < Jt; ++jb) {
    v8f acc = {0.f, 0.f, 0.f, 0.f, 0.f, 0.f, 0.f, 0.f};
    for (int kb = 0; kb < 256; kb += 16) {
      __syncthreads();
      for (int idx = tid; idx < 512; idx += 128) {
        int k = idx >> 5, nn = idx & 31;
        int j = jb * 32 + nn;
        bool ok = j < d;
        float v = dw[(size_t)(kb + k) * d + (ok ? j : 0)];
        Bw[k * HPAD + nn] = ok ? v : 0.0f;
      }
      __syncthreads();
      for (int ks = 0; ks < 4; ++ks) {
        int kk = ks * 4 + 2 * lhi;
        v2f a, bv;
        a.x = hbuf[(wm * 16 + l15) * HPAD + kb + kk];
        a.y = hbuf[(wm * 16 + l15) * HPAD + kb + kk + 1];
        bv.x = Bw[kk * HPAD + wn * 16 + l15];
        bv.y = Bw[(kk + 1) * HPAD + wn * 16 + l15];
        acc = wmma4(a, bv, acc);
      }
    }
    int j = jb * 32 + wn * 16 + l15;
    for (int r = 0; r < 8; ++r) {
      int pm = pBase + wm * 16 + r + 8 * lhi;
      if (pm < P && j < d) {
        size_t idx2 = (size_t)pm * C4 + d + j;
        X[idx2] = X[idx2] - (acc[r] + db[j]);
      }
    }
  }
}

// ---------------- final reduction + epilogue ----------------
__global__ void k_ssq(const float* __restrict__ X, float* ssq, int perBatch) {
  __shared__ float red[256];
  int bpb = perBatch / 1024;                       // float4 per thread, 256 threads
  int b = blockIdx.x / bpb;
  int blk = blockIdx.x % bpb;
  const float4* Xv = (const float4*)(X + (size_t)b * perBatch) + (size_t)blk * 256;
  float4 v = Xv[threadIdx.x];
  red[threadIdx.x] = v.x * v.x + v.y * v.y + v.z * v.z + v.w * v.w;
  __syncthreads();
  for (int st = 128; st > 0; st >>= 1) {
    if (threadIdx.x < st) red[threadIdx.x] += red[threadIdx.x + st];
    __syncthreads();
  }
  if (threadIdx.x == 0) atomicAdd(&ssq[b], red[0]);
}

__global__ void k_final(const float* ssq, const float* ildj, float* out, int B_,
                        float nhalf_log2pi) {
  int b = threadIdx.x;
  if (b < B_) out[b] = -0.5f * ssq[b] - nhalf_log2pi + ildj[0];
}

// ---------------- host driver ----------------
extern "C" void kernel_launch(void* const* d_in, const int* in_sizes, int n_in,
                              void* d_out, int out_size, void* d_ws, size_t ws_size,
                              hipStream_t stream) {
  (void)in_sizes; (void)n_in; (void)out_size; (void)ws_size;
  const float* sample = (const float*)d_in[0];
  float* ws = (float*)d_ws;

  const size_t NTOT = 8ull * 256 * 256 * 3;      // 1,572,864 per tensor
  float* bufA = ws;
  float* bufB = bufA + NTOT;
  float* Asc  = bufB + NTOT;                     // expm scratch (max 768x768)
  float* term = Asc  + 768 * 768;
  float* outm = term + 768 * 768;
  float* tmpm = outm + 768 * 768;
  float* ssq  = tmpm + 768 * 768;                // 8
  float* ildj = ssq + 8;                         // 1
  float* out  = (float*)d_out;

  k_zero<<<1, 32, 0, stream>>>(ssq, 9);

  static const int SARR[7] = {12, 24, 48, 96, 192, 384, 768};
  int Cin = 3, H = 256;
  const float* Xin = sample;

  for (int li = 0; li < 7; ++li) {
    int s = SARR[li];
    int H2 = H / 2, W2 = H / 2;
    int C4 = 4 * Cin;
    int d = s / 2;
    int P = 8 * H2 * W2;

    const float* M  = (const float*)d_in[1 + 6 * li];
    const float* eb = (const float*)d_in[2 + 6 * li];
    const float* cw = (const float*)d_in[3 + 6 * li];
    const float* cb = (const float*)d_in[4 + 6 * li];
    const float* dw = (const float*)d_in[5 + 6 * li];
    const float* db = (const float*)d_in[6 + 6 * li];

    k_ildj<<<1, 32, 0, stream>>>(M, s, (float)H2 * (float)H2, ildj);

    // E = expm(-M): scaling & squaring, all matmuls on WMMA f32
    int n2 = s * s;
    int n4 = n2 / 4;
    int tb  = (n2 + 255) / 256;
    int tb4 = (n4 + 255) / 256;
    k_expm_prep<<<tb, 256, 0, stream>>>(M, Asc, term, outm, s);
    int nt = (s + 31) / 32;
    dim3 gg(nt, nt);
    for (int j = 1; j <= 13; ++j) {
      k_gemm<<<gg, 128, 0, stream>>>(term, Asc, tmpm, s, 1.0f / (float)j);
      k_accum4<<<tb4, 256, 0, stream>>>((float4*)term, (float4*)outm, (const float4*)tmpm, n4);
    }
    for (int q = 0; q < 6; ++q) {
      k_gemm<<<gg, 128, 0, stream>>>(outm, outm, tmpm, s, 1.0f);
      k_copy4<<<tb4, 256, 0, stream>>>((float4*)outm, (const float4*)tmpm, n4);
    }

    float* Xout = (li % 2 == 0) ? bufA : bufB;
    int Pt = (P + 31) / 32;
    dim3 gm(Pt, (s + 31) / 32);
    k_mix<<<gm, 128, 0, stream>>>(Xin, Xout, outm, eb, 8, H2, W2, Cin, s);

    int npass = C4 - s;
    if (npass > 0) {
      long long tot = (long long)P * npass;
      int blocks = (int)((tot + 255) / 256);
      k_passthru<<<blocks, 256, 0, stream>>>(Xin, Xout, 8, H2, W2, Cin, s);
    }

    k_convproj<<<Pt, 128, 0, stream>>>(Xout, cw, cb, dw, db, 8, H2, W2, C4, d);

    Xin = Xout;
    Cin = C4;
    H = H2;
  }

  const int perBatch = 256 * 256 * 3;            // 196,608 (divisible by 1024)
  k_ssq<<<8 * (perBatch / 1024), 256, 0, stream>>>(Xin, ssq, perBatch);
  float nhalf = 0.5f * (float)perBatch * 1.8378770664093453f;
  k_final<<<1, 32, 0, stream>>>(ssq, ildj, out, 8, nhalf);
}